// CRF_343597384390
// MI455X (gfx1250) — compile-verified
//
#include <hip/hip_runtime.h>
#include <hip/hip_bf16.h>

#define CB 512
#define CT 1024
#define CL 48

typedef __attribute__((ext_vector_type(2))) float v2f;
typedef __attribute__((ext_vector_type(8))) float v8f;

__device__ __forceinline__ float gmax16(float v) {
  v = fmaxf(v, __shfl_xor(v, 1));
  v = fmaxf(v, __shfl_xor(v, 2));
  v = fmaxf(v, __shfl_xor(v, 4));
  v = fmaxf(v, __shfl_xor(v, 8));
  return v;
}
__device__ __forceinline__ float gsum16(float v) {
  v += __shfl_xor(v, 1);
  v += __shfl_xor(v, 2);
  v += __shfl_xor(v, 4);
  v += __shfl_xor(v, 8);
  return v;
}

// ---------------------------------------------------------------------------
// Kernel A: per-batch path score (point + transition) and per-(b,t) mask.
// One block per batch element; labels recovered from one-hot y_true.
// ---------------------------------------------------------------------------
__global__ void __launch_bounds__(256)
crf_path_score(const float* __restrict__ y_true, const float* __restrict__ y_pred,
               const float* __restrict__ trans, float* __restrict__ maskbuf,
               float* __restrict__ path) {
  __shared__ int labs[CT];
  __shared__ float red[256];
  const int b = blockIdx.x, tid = threadIdx.x;
  float acc = 0.f;
  for (int t = tid; t < CT; t += 256) {
    const float* row = y_true + ((size_t)b * CT + t) * (CL + 1);
    int lab = CL;
    for (int l = 0; l < CL + 1; ++l)
      if (row[l] > 0.5f) lab = l;
    labs[t] = lab;
    maskbuf[(size_t)b * CT + t] = (lab < CL) ? 1.f : 0.f;
    if (lab < CL) acc += y_pred[((size_t)b * CT + t) * CL + lab];
  }
  __syncthreads();
  for (int t = tid; t < CT - 1; t += 256) {
    int l0 = labs[t], l1 = labs[t + 1];
    if (l0 < CL && l1 < CL) acc += trans[l0 * CL + l1];
  }
  red[tid] = acc;
  __syncthreads();
  for (int s = 128; s > 0; s >>= 1) {
    if (tid < s) red[tid] += red[tid + s];
    __syncthreads();
  }
  if (tid == 0) path[b] = red[0];
}

// ---------------------------------------------------------------------------
// Kernel B: forward recursion. One wave32 per 16 batch rows.
// state (16x48 log-alphas) held in C-layout registers; each step does
// exp-shift -> LDS transpose -> 36x V_WMMA_F32_16X16X4_F32 -> log + emit.
// ---------------------------------------------------------------------------
__global__ void __launch_bounds__(32)
crf_forward(const float* __restrict__ y_pred, const float* __restrict__ trans,
            const float* __restrict__ maskbuf, const float* __restrict__ path,
            float* __restrict__ out) {
  __shared__ float xp[16 * 49];  // padded stride 49: conflict-free transpose
  const int lane = threadIdx.x;
  const int lo = lane & 15;
  const int hi = lane >> 4;
  const int b0 = blockIdx.x * 16;

  // B fragments: E = exp(trans) 48x48, [K-chunk c][N-tile n], resident all T.
  // B layout (4x16 f32): VGPRq, lanes 0-15: K=q, N=lane; lanes 16-31: K=q+2.
  v2f Bfr[12][3];
#pragma unroll
  for (int c = 0; c < 12; ++c)
#pragma unroll
    for (int n = 0; n < 3; ++n) {
      int k0 = 4 * c + 2 * hi;
      int nn = 16 * n + lo;
      Bfr[c][n].x = __expf(trans[(k0 + 0) * CL + nn]);
      Bfr[c][n].y = __expf(trans[(k0 + 1) * CL + nn]);
    }

  // state init = y_pred[:,0,:]; C-layout: st[n][r] = row M=r+8*hi, col 16n+lo
  float st[3][8];
#pragma unroll
  for (int n = 0; n < 3; ++n)
#pragma unroll
    for (int r = 0; r < 8; ++r)
      st[n][r] = y_pred[((size_t)(b0 + r + 8 * hi) * CT) * CL + 16 * n + lo];

  for (int t = 1; t < CT; ++t) {
    if (t + 1 < CT) {  // prefetch next step's emission rows (global_prefetch_b8)
      const float* pf = y_pred + ((size_t)(b0 + lo) * CT + (t + 1)) * CL + hi * 24;
      __builtin_prefetch(pf, 0, 0);
    }
    // per-row max across 48 labels (register max over tiles, shfl over lanes)
    float rm[8];
#pragma unroll
    for (int r = 0; r < 8; ++r)
      rm[r] = gmax16(fmaxf(fmaxf(st[0][r], st[1][r]), st[2][r]));

    // P = exp(state - rowmax) -> LDS [M][K]
#pragma unroll
    for (int n = 0; n < 3; ++n)
#pragma unroll
      for (int r = 0; r < 8; ++r)
        xp[(r + 8 * hi) * 49 + 16 * n + lo] = __expf(st[n][r] - rm[r]);
    __syncthreads();

    // S = P x E : 12 K-chunks x 3 N-tiles of V_WMMA_F32_16X16X4_F32
    v8f acc[3] = {};
#pragma unroll
    for (int c = 0; c < 12; ++c) {
      v2f a;  // A layout (16x4 f32): VGPRq, lanes 0-15: M=lane,K=q; hi: K=q+2
      int k0 = 4 * c + 2 * hi;
      a.x = xp[lo * 49 + k0 + 0];
      a.y = xp[lo * 49 + k0 + 1];
#pragma unroll
      for (int n = 0; n < 3; ++n)
        acc[n] = __builtin_amdgcn_wmma_f32_16x16x4_f32(
            false, a, false, Bfr[c][n], (short)0, acc[n], false, false);
    }
    __syncthreads();

    // state update: new = log(S) + rowmax + emit; masked select per (b,t)
#pragma unroll
    for (int n = 0; n < 3; ++n)
#pragma unroll
      for (int r = 0; r < 8; ++r) {
        size_t bt = (size_t)(b0 + r + 8 * hi) * CT + t;
        float e = y_pred[bt * CL + 16 * n + lo];
        float m = maskbuf[bt];
        float nw = __logf(acc[n][r]) + rm[r] + e;
        st[n][r] = m * nw + (1.f - m) * st[n][r];
      }
  }

  // final: out[b] = logsumexp_j(state[b,j]) - path_score[b]
#pragma unroll
  for (int r = 0; r < 8; ++r) {
    float rm = gmax16(fmaxf(fmaxf(st[0][r], st[1][r]), st[2][r]));
    float s = __expf(st[0][r] - rm) + __expf(st[1][r] - rm) + __expf(st[2][r] - rm);
    s = gsum16(s);
    if (lo == 0) {
      int row = b0 + r + 8 * hi;
      out[row] = __logf(s) + rm - path[row];
    }
  }
}

extern "C" void kernel_launch(void* const* d_in, const int* in_sizes, int n_in,
                              void* d_out, int out_size, void* d_ws, size_t ws_size,
                              hipStream_t stream) {
  const float* y_true = (const float*)d_in[0];  // (B,T,L+1) f32
  const float* y_pred = (const float*)d_in[1];  // (B,T,L)   f32
  const float* trans  = (const float*)d_in[2];  // (L,L)     f32
  float* out = (float*)d_out;                   // (B,1)     f32

  float* maskbuf = (float*)d_ws;                // B*T floats
  float* path = maskbuf + (size_t)CB * CT;      // B floats

  crf_path_score<<<CB, 256, 0, stream>>>(y_true, y_pred, trans, maskbuf, path);
  crf_forward<<<CB / 16, 32, 0, stream>>>(y_pred, trans, maskbuf, path, out);
}